// HypergraphAttentionLayer_31688268710209
// MI455X (gfx1250) — compile-verified
//
#include <hip/hip_runtime.h>
#include <hip/hip_bf16.h>

typedef __attribute__((ext_vector_type(2))) float v2f;
typedef __attribute__((ext_vector_type(8))) float v8f;

#define D_FEAT 64

// Workspace layout (floats): E[M*64] | denom[N] | p[N] | q[M] | eexp[NNZ]
// M lives in device memory (python scalar input), so every kernel derives
// the offsets on-device.
struct Ws {
  float* E;
  float* denom;
  float* p;
  float* q;
  float* eexp;
  int    M;
};

__device__ __forceinline__ Ws make_ws(float* ws, const int* __restrict__ Mptr, int N) {
  Ws w;
  w.M     = Mptr[0];
  w.E     = ws;
  w.denom = w.E + (size_t)w.M * D_FEAT;
  w.p     = w.denom + N;
  w.q     = w.p + N;
  w.eexp  = w.q + w.M;
  return w;
}

// Zero E + denom in workspace, and zero d_out. Grid-stride (M only known on device).
__global__ void k_zero(float* ws, const int* __restrict__ Mptr, int N,
                       float* out, size_t outN) {
  Ws w = make_ws(ws, Mptr, N);
  size_t zlen   = (size_t)w.M * D_FEAT + (size_t)N; // E then denom (contiguous)
  size_t stride = (size_t)gridDim.x * blockDim.x;
  size_t t0     = (size_t)blockIdx.x * blockDim.x + threadIdx.x;
  for (size_t i = t0; i < zlen; i += stride) ws[i] = 0.0f;
  for (size_t i = t0; i < outN; i += stride) out[i] = 0.0f;
}

// Pass 1: E[edge_idx[z], :] += values[z] * X[node_idx[z], :]
// One thread per (z, d) element; f32 atomics land in L2 (E = 5 MB, L2-resident).
__global__ void k_scatter_E(const float* __restrict__ X, const float* __restrict__ values,
                            const int* __restrict__ node_idx, const int* __restrict__ edge_idx,
                            float* __restrict__ E, int nnz) {
  size_t t = (size_t)blockIdx.x * blockDim.x + threadIdx.x;
  size_t z = t >> 6;
  int    d = (int)(t & 63);
  if (z >= (size_t)nnz) return;
  int   n = node_idx[z];
  int   m = edge_idx[z];
  float v = values[z];
  atomicAdd(&E[(size_t)m * D_FEAT + d], v * X[(size_t)n * D_FEAT + d]);
}

// Dense GEMV via V_WMMA_F32_16X16X4_F32 (full f32 precision).
// which==0: p = X @ a[0:64]   (rows = N)
// which==1: q = E @ a[64:128] (rows = M, read on device)
//
// Per wave: 16 rows. A fragment (16x4 f32): lane L -> row = L%16, K = (L/16)*2 + v.
// B fragment (4x16 f32): lane L -> K = (L/16)*2 + v, col = L%16; we replicate the
// vector across all 16 columns so every column of D holds the dot product.
// D (16x16 f32, 8 VGPRs): VGPR r, lane L -> row = r + 8*(L/16); extract at col 0
// (lanes 0 and 16).
__global__ void k_gemv_wmma(const float* __restrict__ X, const float* __restrict__ a,
                            float* ws, const int* __restrict__ Mptr, int N, int which) {
  Ws w = make_ws(ws, Mptr, N);
  const float* mat  = which ? w.E : X;
  const float* vec  = a + (which ? D_FEAT : 0);
  float*       outv = which ? w.q : w.p;
  int          rows = which ? w.M : N;

  int lane  = threadIdx.x & 31;
  int wave  = threadIdx.x >> 5;
  int wavesPerBlock = blockDim.x >> 5;
  int waveId   = blockIdx.x * wavesPerBlock + wave;
  int numWaves = gridDim.x * wavesPerBlock;

  int m     = lane & 15;
  int khalf = lane >> 4;   // 0 or 1

  // Preload the 16 B fragments (the 64-entry vector, broadcast to all columns).
  v2f B[16];
#pragma unroll
  for (int chunk = 0; chunk < 16; ++chunk) {
    int kb = chunk * 4 + khalf * 2;
    B[chunk].x = vec[kb];
    B[chunk].y = vec[kb + 1];
  }

  // Wave-uniform loop bounds: EXEC stays all-1s around the WMMAs.
  for (int rowBase = waveId * 16; rowBase < rows; rowBase += numWaves * 16) {
    int r = rowBase + m;
    if (r > rows - 1) r = rows - 1;          // tail clamp (reads only; stores masked)
    const float* rowPtr = mat + (size_t)r * D_FEAT;
    v8f c = {};
#pragma unroll
    for (int chunk = 0; chunk < 16; ++chunk) {
      int kb = chunk * 4 + khalf * 2;
      v2f A;
      A.x = rowPtr[kb];
      A.y = rowPtr[kb + 1];
      c = __builtin_amdgcn_wmma_f32_16x16x4_f32(
          /*neg_a=*/false, A, /*neg_b=*/false, B[chunk],
          /*c_mod=*/(short)0, c, /*reuse_a=*/false, /*reuse_b=*/false);
    }
    if (m == 0) {
      int base = rowBase + 8 * khalf;
#pragma unroll
      for (int rr = 0; rr < 8; ++rr) {
        if (base + rr < rows) outv[base + rr] = c[rr];
      }
    }
  }
}

// Pass 2: e_exp[z] = exp(p[n] + q[m]); denom[n] += e_exp[z]
__global__ void k_denom(const int* __restrict__ node_idx, const int* __restrict__ edge_idx,
                        float* ws, const int* __restrict__ Mptr, int N, int nnz) {
  Ws w = make_ws(ws, Mptr, N);
  int z = blockIdx.x * blockDim.x + threadIdx.x;
  if (z >= nnz) return;
  int   n = node_idx[z];
  int   m = edge_idx[z];
  float e = __expf(w.p[n] + w.q[m]);
  w.eexp[z] = e;
  atomicAdd(&w.denom[n], e);
}

// Pass 3: X_out[n, :] += (e_exp[z] / (denom[n] + 1e-16)) * E[m, :]
__global__ void k_gather_out(const int* __restrict__ node_idx, const int* __restrict__ edge_idx,
                             float* ws, const int* __restrict__ Mptr, int N, int nnz,
                             float* __restrict__ out) {
  Ws w = make_ws(ws, Mptr, N);
  size_t t = (size_t)blockIdx.x * blockDim.x + threadIdx.x;
  size_t z = t >> 6;
  int    d = (int)(t & 63);
  if (z >= (size_t)nnz) return;
  int   n     = node_idx[z];
  int   m     = edge_idx[z];
  float alpha = w.eexp[z] / (w.denom[n] + 1e-16f);
  atomicAdd(&out[(size_t)n * D_FEAT + d], alpha * w.E[(size_t)m * D_FEAT + d]);
}

extern "C" void kernel_launch(void* const* d_in, const int* in_sizes, int n_in,
                              void* d_out, int out_size, void* d_ws, size_t ws_size,
                              hipStream_t stream) {
  const float* X        = (const float*)d_in[0];
  const float* values   = (const float*)d_in[1];
  const float* a        = (const float*)d_in[2];
  const int*   node_idx = (const int*)d_in[3];
  const int*   edge_idx = (const int*)d_in[4];
  const int*   Mptr     = (const int*)d_in[5];   // num_hyperedges (device scalar)
  float*       out      = (float*)d_out;
  float*       ws       = (float*)d_ws;

  const int N   = in_sizes[0] / D_FEAT;
  const int NNZ = in_sizes[1];

  // Zero E + denom + output (grid-stride; covers M*64+N on device side).
  int nzB = (int)(((size_t)out_size + 255) / 256);
  if (nzB < 1) nzB = 1;
  k_zero<<<nzB, 256, 0, stream>>>(ws, Mptr, N, out, (size_t)out_size);

  // Scatter into E.
  size_t scT = (size_t)NNZ * D_FEAT;
  int scB = (int)((scT + 255) / 256);
  k_scatter_E<<<scB, 256, 0, stream>>>(X, values, node_idx, edge_idx, ws, NNZ);

  // p = X @ a_x  (independent of E), then q = E @ a_e (needs scatter done).
  k_gemv_wmma<<<1024, 256, 0, stream>>>(X, a, ws, Mptr, N, /*which=*/0);
  k_gemv_wmma<<<1024, 256, 0, stream>>>(X, a, ws, Mptr, N, /*which=*/1);

  // Softmax denominator per node.
  k_denom<<<(NNZ + 255) / 256, 256, 0, stream>>>(node_idx, edge_idx, ws, Mptr, N, NNZ);

  // Weighted scatter to output.
  k_gather_out<<<scB, 256, 0, stream>>>(node_idx, edge_idx, ws, Mptr, N, NNZ, out);
}